// AtomToEdgeLayer_21191368639072
// MI455X (gfx1250) — compile-verified
//
#include <hip/hip_runtime.h>
#include <hip/hip_bf16.h>

typedef __attribute__((ext_vector_type(2))) float v2f;
typedef __attribute__((ext_vector_type(8))) float v8f;

#define D_EDGE 128
#define D_RBF  64

__device__ __forceinline__ float ssp_ln2(float x) {
    // softplus(x) - ln2, numerically stable
    float ax = fabsf(x);
    float t  = __logf(1.0f + __expf(-ax));
    return fmaxf(x, 0.0f) + t - 0.6931471805599453f;
}

// ---------------------------------------------------------------------------
// Kernel 1: fold weights into K-paired (v2f) layouts for WMMA B-operands.
//   Wc[n][k]  = sum_d W_cat[n][d] * W_rbf[d][k]        (k < 64)
//   bc[n]     = b_cat[n] + sum_d W_cat[n][d] * b_rbf[d]
//   WcP[(k/2)*128 + n] = { Wc[n][k], Wc[n][k+1] }       (k even)
//   W2P[(k/2)*128 + n] = { W_cat[n][128+k], W_cat[n][128+k+1] }
//   W3P[(k/2)*128 + n] = { W_cat[n][256+k], W_cat[n][256+k+1] }
// grid 128 blocks (n), 128 threads (k)
// ---------------------------------------------------------------------------
__global__ void fold_weights_kernel(const float* __restrict__ W_rbf,
                                    const float* __restrict__ b_rbf,
                                    const float* __restrict__ W_cat,
                                    const float* __restrict__ b_cat,
                                    float* __restrict__ WcP,
                                    float* __restrict__ bc,
                                    float* __restrict__ W2P,
                                    float* __restrict__ W3P) {
    const int n = blockIdx.x;
    const int k = threadIdx.x;
    const float* wrow = W_cat + n * (3 * D_EDGE);
    // paired index: element (n, k) lives at [(k>>1)*128 + n]*2 + (k&1)
    const int pidx = ((k >> 1) * D_EDGE + n) * 2 + (k & 1);
    if (k < D_RBF) {
        float s = 0.0f;
        #pragma unroll 8
        for (int d = 0; d < D_EDGE; ++d)
            s = fmaf(wrow[d], W_rbf[d * D_RBF + k], s);
        WcP[pidx] = s;
    }
    W2P[pidx] = wrow[D_EDGE + k];
    W3P[pidx] = wrow[2 * D_EDGE + k];
    if (k == 0) {
        float s = b_cat[n];
        #pragma unroll 8
        for (int d = 0; d < D_EDGE; ++d)
            s = fmaf(wrow[d], b_rbf[d], s);
        bc[n] = s;
    }
}

// ---------------------------------------------------------------------------
// Kernel 2: per-node projections P2 = vi @ W2^T, P3 = vi @ W3^T via WMMA f32.
// grid.x = ceil(N/16) node tiles, grid.y = {0:P2, 1:P3}; 256 threads = 8 waves,
// wave w owns output columns [16w, 16w+16).
// ---------------------------------------------------------------------------
__global__ __launch_bounds__(256) void node_proj_kernel(
        const float* __restrict__ vi,
        const float* __restrict__ W2P, const float* __restrict__ W3P,
        float* __restrict__ P2, float* __restrict__ P3, int N) {
    const v2f* __restrict__ Wp = (const v2f*)(blockIdx.y ? W3P : W2P);
    float*     __restrict__ P  = blockIdx.y ? P3 : P2;

    const int wave = threadIdx.x >> 5;
    const int lane = threadIdx.x & 31;
    const int half = lane >> 4;        // K half-select (A/B layout)
    const int m    = lane & 15;        // A row / B col within tile
    const int n    = wave * 16 + m;    // output column

    const int  row  = blockIdx.x * 16 + m;
    const int  rowc = row < N ? row : N - 1;
    const v2f* arow = (const v2f*)(vi + (size_t)rowc * D_EDGE);

    v8f acc = {};
    #pragma unroll
    for (int k0 = 0; k0 < D_EDGE; k0 += 4) {
        const int kp = (k0 >> 1) + half;          // pair index (k = 2*kp)
        v2f a = arow[kp];
        v2f b = Wp[kp * D_EDGE + n];
        acc = __builtin_amdgcn_wmma_f32_16x16x4_f32(
                  false, a, false, b, (short)0, acc, false, false);
    }

    #pragma unroll
    for (int r = 0; r < 8; ++r) {
        const int node = blockIdx.x * 16 + r + 8 * half;  // C/D layout: M = r + 8*half
        if (node < N) P[(size_t)node * D_EDGE + n] = acc[r];
    }
}

// ---------------------------------------------------------------------------
// Kernel 3: per-edge: out[e] = ssp( rbf[e] @ Wc^T + bc + P2[j(e)] + P3[i(e)] )
// 16 edges per block, 8 waves cover 128 output cols. K=64 -> 16 WMMA ops.
// Output stored non-temporal (write-once 328 MB stream; keep L2 for P2/P3).
// ---------------------------------------------------------------------------
__global__ __launch_bounds__(256) void edge_msg_kernel(
        const float* __restrict__ rbf,
        const long long* __restrict__ edge_index,   // (2, E) int64
        const float* __restrict__ WcP,
        const float* __restrict__ bc,
        const float* __restrict__ P2,
        const float* __restrict__ P3,
        float* __restrict__ out, int E) {
    const v2f* __restrict__ Wp = (const v2f*)WcP;

    const int wave = threadIdx.x >> 5;
    const int lane = threadIdx.x & 31;
    const int half = lane >> 4;
    const int m    = lane & 15;
    const int n    = wave * 16 + m;

    const long long e  = (long long)blockIdx.x * 16 + m;
    const long long ec = e < E ? e : (long long)E - 1;
    const v2f* arow = (const v2f*)(rbf + ec * D_RBF);

    v8f acc = {};
    #pragma unroll
    for (int k0 = 0; k0 < D_RBF; k0 += 4) {
        const int kp = (k0 >> 1) + half;
        v2f a = arow[kp];
        v2f b = Wp[kp * D_EDGE + n];
        acc = __builtin_amdgcn_wmma_f32_16x16x4_f32(
                  false, a, false, b, (short)0, acc, false, false);
    }

    const float bias = bc[n];
    #pragma unroll
    for (int r = 0; r < 8; ++r) {
        const long long er = (long long)blockIdx.x * 16 + r + 8 * half;
        if (er < E) {
            const long long i = edge_index[er];        // row 0: h_i source
            const long long j = edge_index[E + er];    // row 1: h_j source
            float y = acc[r] + bias
                    + P2[j * D_EDGE + n]
                    + P3[i * D_EDGE + n];
            __builtin_nontemporal_store(ssp_ln2(y), &out[er * D_EDGE + n]);
        }
    }
}

// ---------------------------------------------------------------------------
extern "C" void kernel_launch(void* const* d_in, const int* in_sizes, int n_in,
                              void* d_out, int out_size, void* d_ws, size_t ws_size,
                              hipStream_t stream) {
    const float*     vi    = (const float*)d_in[0];
    const float*     rbf   = (const float*)d_in[1];
    const float*     W_rbf = (const float*)d_in[2];
    const float*     b_rbf = (const float*)d_in[3];
    const float*     W_cat = (const float*)d_in[4];
    const float*     b_cat = (const float*)d_in[5];
    const long long* eidx  = (const long long*)d_in[6];
    float*           out   = (float*)d_out;

    const int N = in_sizes[0] / D_EDGE;   // 50000
    const int E = in_sizes[1] / D_RBF;    // 640000

    // Workspace layout (floats)
    float* ws  = (float*)d_ws;
    float* WcP = ws;                               // 64*128   = 8192
    float* bc  = WcP + D_RBF * D_EDGE;             // 128
    float* W2P = bc  + D_EDGE;                     // 128*128  = 16384
    float* W3P = W2P + D_EDGE * D_EDGE;            // 128*128  = 16384
    float* P2  = W3P + D_EDGE * D_EDGE;            // N*128
    float* P3  = P2  + (size_t)N * D_EDGE;         // N*128

    fold_weights_kernel<<<D_EDGE, D_EDGE, 0, stream>>>(
        W_rbf, b_rbf, W_cat, b_cat, WcP, bc, W2P, W3P);

    dim3 gridP((N + 15) / 16, 2);
    node_proj_kernel<<<gridP, 256, 0, stream>>>(vi, W2P, W3P, P2, P3, N);

    edge_msg_kernel<<<(E + 15) / 16, 256, 0, stream>>>(
        rbf, eidx, WcP, bc, P2, P3, out, E);
}